// NAM_32229434589251
// MI455X (gfx1250) — compile-verified
//
#include <hip/hip_runtime.h>

// ---------------------------------------------------------------------------
// Fused NAM forward for MI455X (gfx1250, wave32, WMMA).
//   out = concat_f( relu(x[:,f] (x) w1[f] + b1[f]) @ w2[f] + b2[f] ) @ wout^T + bout
// Fully fused (feats never touch HBM); contraction on v_wmma_f32_16x16x4_f32.
// B panels stored in LDS in a fragment-native layout: element (k,n) at
//   q*STRIDE + n*4 + (k&3),  q = k>>2
// so every WMMA B fragment is ONE aligned ds_load_b64 into an even VGPR pair,
// hitting all 64 LDS banks exactly once (no repack movs, no conflicts).
// ---------------------------------------------------------------------------

typedef __attribute__((ext_vector_type(2))) float v2f;
typedef __attribute__((ext_vector_type(8))) float v8f;

#define B_TOT   8192
#define F_TOT   64
#define H_DIM   128
#define C_DIM   32
#define FH      (F_TOT * H_DIM)
#define BTILE   128                 // batch rows per workgroup
#define FSPLIT  4                   // feature groups (partials via f32 atomics)
#define FPB     (F_TOT / FSPLIT)    // 16 features per block
#define NTHREADS 256                // 8 waves of 32
#define HP      132                 // hs row stride (A panel), 64-bank friendly
#define W2S     512                 // w2 chunk stride: 128 cols * 4 slots
#define WOS     132                 // wout chunk stride: 32 cols * 4 slots + pad

// LDS (floats): hs[128][132] | w2q[32][512] | woutq[32][132] | w1s,b1s,b2s[128] | xs[128]
#define SMEM_FLOATS (BTILE*HP + 32*W2S + 32*WOS + 3*H_DIM + BTILE)

__global__ void nam_init_out(const float* __restrict__ bout,
                             float* __restrict__ out, int n) {
    int i = blockIdx.x * blockDim.x + threadIdx.x;
    if (i < n) out[i] = bout[i & (C_DIM - 1)];
}

__global__ void __launch_bounds__(NTHREADS)
nam_fused_wmma(const float* __restrict__ x,
               const float* __restrict__ w1,
               const float* __restrict__ b1,
               const float* __restrict__ w2,
               const float* __restrict__ b2,
               const float* __restrict__ wout,
               float* __restrict__ out) {
    extern __shared__ float smem[];
    float* hs    = smem;                       // [BTILE][HP]   A panel / feats
    float* w2q   = hs    + BTILE * HP;         // [32][W2S]     GEMM2 B panel
    float* woutq = w2q   + 32 * W2S;           // [32][WOS]     head  B panel
    float* w1s   = woutq + 32 * WOS;           // [H]
    float* b1s   = w1s   + H_DIM;              // [H]
    float* b2s   = b1s   + H_DIM;              // [H]
    float* xs    = b2s   + H_DIM;              // [BTILE]

    const int t    = threadIdx.x;
    const int wave = t >> 5;
    const int lane = t & 31;
    const int half = lane >> 4;   // 0: lanes 0-15 (K 0,1), 1: lanes 16-31 (K 2,3)
    const int lm   = lane & 15;

    const int btile = blockIdx.x % (B_TOT / BTILE);
    const int fg    = blockIdx.x / (B_TOT / BTILE);
    const int b0    = btile * BTILE;
    const int m0    = wave * 16;  // wave-private 16-row window of the tile

    const v8f vzero = {};
    v8f headacc[2];               // two 16x16 tiles over C=32, persistent over f
    headacc[0] = vzero;
    headacc[1] = vzero;

    for (int ff = 0; ff < FPB; ++ff) {
        const int f = fg * FPB + ff;

        // Pull next feature's w2 panel toward this WGP while we compute.
        if (ff + 1 < FPB) {
            const float* nxt = w2 + (size_t)(f + 1) * H_DIM * H_DIM;
            __builtin_prefetch(nxt + t * 64, 0, 3);
        }

        __syncthreads();  // previous iteration's readers of B panels are done

        // ---- stage per-feature vectors ----
        if (t < H_DIM) {
            w1s[t] = w1[f * H_DIM + t];
            b1s[t] = b1[f * H_DIM + t];
            b2s[t] = b2[f * H_DIM + t];
        } else {
            const int r = t - H_DIM;                    // 0..127
            xs[r] = x[(size_t)(b0 + r) * F_TOT + f];
        }

        // ---- stage w2[f] into fragment-native layout (b64 stores) ----
        // element (kc, n) -> w2q[(kc>>2)*W2S + n*4 + (kc&3)]
        {
            const float* w2f = w2 + (size_t)f * H_DIM * H_DIM;
            #pragma unroll 4
            for (int i = t; i < (H_DIM / 2) * H_DIM; i += NTHREADS) {
                const int n  = i & 127;
                const int kp = i >> 7;          // 0..63, kc = 2*kp
                v2f v;
                v.x = w2f[(2 * kp) * H_DIM + n];
                v.y = w2f[(2 * kp) * H_DIM + H_DIM + n];
                *(v2f*)&w2q[(kp >> 1) * W2S + n * 4 + 2 * (kp & 1)] = v;
            }
        }
        // ---- stage wout K-slice: element (kc, c) = wout[c, f*H + kc] ----
        #pragma unroll 2
        for (int i = t; i < (H_DIM / 2) * C_DIM; i += NTHREADS) {
            const int kp = i & 63;              // kc = 2*kp (contiguous pair)
            const int c  = i >> 6;              // 0..31
            v2f g = *(const v2f*)&wout[(size_t)c * FH + f * H_DIM + 2 * kp];
            *(v2f*)&woutq[(kp >> 1) * WOS + c * 4 + 2 * (kp & 1)] = g;
        }
        __syncthreads();

        // ---- layer 1 (rank-1 + relu) into wave-private rows of hs ----
        #pragma unroll 4
        for (int i = lane; i < 16 * H_DIM; i += 32) {
            const int r = i >> 7, cc = i & 127;
            const float v = fmaf(xs[m0 + r], w1s[cc], b1s[cc]);
            hs[(m0 + r) * HP + cc] = v > 0.0f ? v : 0.0f;
        }
        // rows m0..m0+15 written & read only by this wave -> no barrier needed

        // ---- GEMM2: feats(16x128) = hs(16x128) @ w2[f](128x128) ----
        v8f acc[8];
        #pragma unroll
        for (int nt = 0; nt < 8; ++nt) acc[nt] = vzero;

        const int a_row = m0 + lm;
        for (int k0 = 0; k0 < H_DIM; k0 += 4) {
            // A frag: one aligned b64 (row-contiguous)
            v2f a = *(const v2f*)&hs[a_row * HP + k0 + 2 * half];
            const float* bq = &w2q[(k0 >> 2) * W2S + 2 * half];
            #pragma unroll
            for (int nt = 0; nt < 8; ++nt) {
                // B frag: one aligned b64, conflict-free across the wave
                v2f bf = *(const v2f*)&bq[(nt * 16 + lm) * 4];
                acc[nt] = __builtin_amdgcn_wmma_f32_16x16x4_f32(
                    false, a, false, bf, (short)0, acc[nt], false, false);
            }
        }

        // ---- feats (+b2) back into the same wave-private rows ----
        // D layout: VGPR r -> row m0 + r + 8*half, col = nt*16 + lm
        #pragma unroll
        for (int nt = 0; nt < 8; ++nt) {
            #pragma unroll
            for (int r = 0; r < 8; ++r) {
                hs[(m0 + r + 8 * half) * HP + nt * 16 + lm] =
                    acc[nt][r] + b2s[nt * 16 + lm];
            }
        }

        // ---- head: out(16x32) += feats(16x128) @ wout_slice(128x32) ----
        for (int k0 = 0; k0 < H_DIM; k0 += 4) {
            v2f a = *(const v2f*)&hs[a_row * HP + k0 + 2 * half];
            const float* bq = &woutq[(k0 >> 2) * WOS + 2 * half];
            #pragma unroll
            for (int nt = 0; nt < 2; ++nt) {
                v2f bf = *(const v2f*)&bq[(nt * 16 + lm) * 4];
                headacc[nt] = __builtin_amdgcn_wmma_f32_16x16x4_f32(
                    false, a, false, bf, (short)0, headacc[nt], false, false);
            }
        }
    }

    // ---- combine feature-group partials into out (pre-seeded with bout) ----
    #pragma unroll
    for (int nt = 0; nt < 2; ++nt) {
        #pragma unroll
        for (int r = 0; r < 8; ++r) {
            atomicAdd(&out[(size_t)(b0 + m0 + r + 8 * half) * C_DIM +
                           nt * 16 + lm],
                      headacc[nt][r]);
        }
    }
}

extern "C" void kernel_launch(void* const* d_in, const int* in_sizes, int n_in,
                              void* d_out, int out_size, void* d_ws, size_t ws_size,
                              hipStream_t stream) {
    const float* x    = (const float*)d_in[0];
    const float* w1   = (const float*)d_in[1];
    const float* b1   = (const float*)d_in[2];
    const float* w2   = (const float*)d_in[3];
    const float* b2   = (const float*)d_in[4];
    const float* wout = (const float*)d_in[5];
    const float* bout = (const float*)d_in[6];
    float* out = (float*)d_out;

    const int n_out = B_TOT * C_DIM;
    nam_init_out<<<(n_out + 255) / 256, 256, 0, stream>>>(bout, out, n_out);

    const size_t smem_bytes = (size_t)SMEM_FLOATS * sizeof(float);
    const int nblocks = (B_TOT / BTILE) * FSPLIT;   // 256 workgroups
    nam_fused_wmma<<<nblocks, NTHREADS, smem_bytes, stream>>>(
        x, w1, b1, w2, b2, wout, out);
}